// ResidualVectorQuantizer_7000796692960
// MI455X (gfx1250) — compile-verified
//
#include <hip/hip_runtime.h>

typedef __attribute__((ext_vector_type(2))) float v2f;
typedef __attribute__((ext_vector_type(4))) float v4f;
typedef __attribute__((ext_vector_type(8))) float v8f;
typedef __attribute__((ext_vector_type(4))) unsigned u4v;
typedef __attribute__((ext_vector_type(4))) int i4v;
typedef __attribute__((ext_vector_type(8))) int i8v;

#define B_    16
#define D_    256
#define T_    4096
#define NQ    8
#define BINS  1024
#define TM    128            // rows (t positions) per workgroup
#define RS    260            // padded LDS row stride in floats (260 % 64 == 4 -> conflict-free frag reads)
#define CHUNK 64             // codebook bins per LDS chunk (double buffered)
#define NCHUNK (BINS / CHUNK)
#define NT    (CHUNK / 16)   // 4 N-tiles per chunk per wave

// ---- TDM: DMA one codebook chunk (CHUNK rows x 256 f32, row stride 256 f32) into LDS,
//      with hardware padding: pad_interval=7 (every 256 DWORDs) + pad_amount=3 (4 DWORDs)
//      -> LDS row stride 260 floats, matching RS. Tracked with TENSORcnt.
__device__ __forceinline__ void tdm_load_chunk(const float* gsrc, const float* ldsDst) {
    unsigned long long ga = (unsigned long long)(size_t)gsrc;
    u4v g0;
    g0[0] = 1u;                                              // count=1 (valid user descriptor)
    g0[1] = (unsigned)(size_t)ldsDst;                        // lds_addr (bytes)
    g0[2] = (unsigned)ga;                                    // global_addr[31:0]
    g0[3] = (unsigned)((ga >> 32) & 0x01ffffffull)           // global_addr[56:32]
          | (2u << 30);                                      // type=2 ("image")
    i8v g1;
    g1[0] = (int)((2u << 16)                                 // data_size = 4 bytes
                | (1u << 20)                                 // pad_enable
                | (7u << 22)                                 // pad_interval: 256 DWORDs
                | (3u << 25));                               // pad_amount: 4 DWORDs
    g1[1] = (int)(256u << 16);                               // tensor_dim0 = 256 (lo16)
    g1[2] = (int)(1024u << 16);                              // dim0 hi=0 | tensor_dim1 = 1024 (lo16)
    g1[3] = (int)(256u << 16);                               // dim1 hi=0 | tile_dim0 = 256
    g1[4] = (int)CHUNK;                                      // tile_dim1 = 64, tile_dim2 = 0
    g1[5] = 256;                                             // tensor_dim0_stride = 256
    g1[6] = 0;
    g1[7] = 0;
    i4v z4 = {0, 0, 0, 0};
#if defined(__clang_major__) && (__clang_major__ >= 23)
    i8v z8 = {0, 0, 0, 0, 0, 0, 0, 0};
    __builtin_amdgcn_tensor_load_to_lds(g0, g1, z4, z4, z8, 0);
#else
    __builtin_amdgcn_tensor_load_to_lds(g0, g1, z4, z4, 0);
#endif
}

__global__ __launch_bounds__(256)
void rvq_fused_kernel(const float* __restrict__ x,
                      const float* __restrict__ cb,
                      float* __restrict__ out_q,
                      float* __restrict__ out_codes)
{
    extern __shared__ float smem[];
    float* Rs    = smem;                          // TM x RS residual tile
    float* Bs    = smem + TM * RS;                // 2 x (CHUNK x RS) codebook double buffer
    float* Esq   = Bs + 2 * CHUNK * RS;           // 2 x CHUNK per-bin |e|^2
    float* Epart = Esq + 2 * CHUNK;               // 4 x CHUNK partial sums

    const int tid  = threadIdx.x;
    const int lane = tid & 31;                    // wave32
    const int wave = tid >> 5;                    // 0..7

    const int blk = blockIdx.x;                   // 512 blocks
    const int b   = blk >> 5;                     // T_/TM = 32 tiles per batch entry
    const int t0  = (blk & 31) * TM;

    // ---- load residual tile (transpose): Rs[tl][d] = x[b][d][t0+tl], coalesced in t
    {
        const int tl = tid & (TM - 1);
        const int d0 = tid >> 7;                  // 0..1
        const float* xb = x + (size_t)b * D_ * T_ + t0 + tl;
        for (int d = d0; d < D_; d += 2)
            Rs[tl * RS + d] = xb[(size_t)d * T_];
    }

    const int rowM = wave * 16;                   // this wave's 16 rows
    const int am   = lane & 15;                   // A-row / B-col index within 16
    const int kb   = (lane >> 4) * 2;             // K sub-offset per ISA 32-bit A/B layout

    for (int q = 0; q < NQ; ++q) {
        const float* cbq = cb + (size_t)q * BINS * D_;

        __syncthreads();                          // all reads of Bs from previous stage done
        if (wave == 0)                            // prefetch chunk 0 via TDM (EXEC-independent)
            tdm_load_chunk(cbq, Bs);

        float best[8];
        int   bidx[8];
        #pragma unroll
        for (int i = 0; i < 8; ++i) { best[i] = -3.4e38f; bidx[i] = 0; }

        for (int nc = 0; nc < NCHUNK; ++nc) {
            const int p = nc & 1;
            float* Bp = Bs + p * (CHUNK * RS);

            if (wave == 0)
                __builtin_amdgcn_s_wait_tensorcnt(0);   // chunk nc has landed in LDS
            __syncthreads();

            // ---- e_sq for this chunk: 4-way split, float4 LDS reads
            {
                const int n    = tid & (CHUNK - 1);
                const int part = tid >> 6;              // 0..3
                const float* r = Bp + n * RS + part * 64;
                float s = 0.f;
                #pragma unroll
                for (int d = 0; d < 64; d += 4) {
                    v4f v = *(const v4f*)(r + d);
                    s += v[0]*v[0] + v[1]*v[1] + v[2]*v[2] + v[3]*v[3];
                }
                Epart[part * CHUNK + n] = s;
            }
            __syncthreads();
            if (tid < CHUNK)
                Esq[p * CHUNK + tid] = Epart[tid] + Epart[CHUNK + tid]
                                     + Epart[2 * CHUNK + tid] + Epart[3 * CHUNK + tid];
            __syncthreads();

            // ---- kick off DMA of next chunk into the other buffer (overlaps WMMA below)
            if (nc + 1 < NCHUNK && wave == 0)
                tdm_load_chunk(cbq + (size_t)(nc + 1) * CHUNK * D_, Bs + (p ^ 1) * (CHUNK * RS));

            // ---- dots: 16 rows x 64 bins via v_wmma_f32_16x16x4_f32, K=256
            v8f acc[NT];
            {
                v8f zero = {0.f, 0.f, 0.f, 0.f, 0.f, 0.f, 0.f, 0.f};
                #pragma unroll
                for (int nt = 0; nt < NT; ++nt) acc[nt] = zero;
            }
            const float* Arow = Rs + (rowM + am) * RS;
            for (int k = 0; k < D_; k += 4) {
                v2f a = *(const v2f*)(Arow + k + kb);      // A reused across N-tiles
                #pragma unroll
                for (int nt = 0; nt < NT; ++nt) {
                    v2f bf = *(const v2f*)(Bp + (nt * 16 + am) * RS + k + kb);
                    acc[nt] = __builtin_amdgcn_wmma_f32_16x16x4_f32(
                        false, a, false, bf, (short)0, acc[nt], false, false);
                }
            }

            // ---- per-lane argmax of score = 2*dot - e_sq (x_sq is per-row constant)
            #pragma unroll
            for (int nt = 0; nt < NT; ++nt) {
                int   nl = nt * 16 + am;
                int   ng = nc * CHUNK + nl;
                float es = Esq[p * CHUNK + nl];
                #pragma unroll
                for (int i = 0; i < 8; ++i) {
                    float sc = 2.f * acc[nt][i] - es;
                    if (sc > best[i]) { best[i] = sc; bidx[i] = ng; }   // strict > keeps first idx
                }
            }
        }

        // ---- argmax across the 16 column-lanes (both wave halves independent)
        #pragma unroll
        for (int i = 0; i < 8; ++i) {
            float v  = best[i];
            int   ix = bidx[i];
            #pragma unroll
            for (int off = 8; off >= 1; off >>= 1) {
                float ov = __shfl_xor(v, off, 16);
                int   oi = __shfl_xor(ix, off, 16);
                if (ov > v || (ov == v && oi < ix)) { v = ov; ix = oi; }
            }
            best[i] = v; bidx[i] = ix;
        }

        // ---- emit codes + residual -= embed[idx] (wave-local rows)
        #pragma unroll
        for (int r = 0; r < 16; ++r) {
            int idx_r = __shfl(bidx[r & 7], (r >> 3) * 16, 32);
            if (lane == 0)
                out_codes[(size_t)q * (B_ * T_) + (size_t)b * T_ + t0 + rowM + r] = (float)idx_r;
            const v4f* e  = (const v4f*)(cbq + (size_t)idx_r * D_);
            float*     Rr = Rs + (rowM + r) * RS;
            #pragma unroll
            for (int j = 0; j < 2; ++j) {
                int d4 = j * 32 + lane;               // 64 float4 = 256 floats
                v4f ev = e[d4];
                v4f rv = *(v4f*)(Rr + d4 * 4);
                rv = rv - ev;
                *(v4f*)(Rr + d4 * 4) = rv;
            }
        }
    }

    __syncthreads();  // final residual visible across waves
    // ---- quantized = x - residual_final (transpose store, coalesced in t)
    {
        const int tl = tid & (TM - 1);
        const int d0 = tid >> 7;
        const float* xb = x     + (size_t)b * D_ * T_ + t0 + tl;
        float*       ob = out_q + (size_t)b * D_ * T_ + t0 + tl;
        for (int d = d0; d < D_; d += 2)
            ob[(size_t)d * T_] = xb[(size_t)d * T_] - Rs[tl * RS + d];
    }
}

extern "C" void kernel_launch(void* const* d_in, const int* in_sizes, int n_in,
                              void* d_out, int out_size, void* d_ws, size_t ws_size,
                              hipStream_t stream) {
    (void)in_sizes; (void)n_in; (void)d_ws; (void)ws_size; (void)out_size;
    const float* x  = (const float*)d_in[0];   // [16, 256, 4096] f32
    const float* cb = (const float*)d_in[1];   // [8, 1024, 256] f32
    float* out_q     = (float*)d_out;                          // [16,256,4096]
    float* out_codes = out_q + (size_t)B_ * D_ * T_;           // [8,16,4096] as float values

    size_t smem = (size_t)(TM * RS + 2 * CHUNK * RS + 2 * CHUNK + 4 * CHUNK) * sizeof(float);
    rvq_fused_kernel<<<(B_ * T_) / TM, 256, smem, stream>>>(x, cb, out_q, out_codes);
}